// VQC_41197326303720
// MI455X (gfx1250) — compile-verified
//
#include <hip/hip_runtime.h>
#include <hip/hip_bf16.h>

typedef __attribute__((ext_vector_type(2))) float v2f;
typedef __attribute__((ext_vector_type(8))) float v8f;

#define NQ 4
#define NL 8
#define DIM 16
#define WPB 8   // waves per block (256 threads)
#define TPW 8   // 16-sample tiles per wave

// ---------------------------------------------------------------------------
// Prologue: build the batch-invariant 16x16 circuit unitary M (8 layers of
// Rot + CNOT ring), then H = M^dagger * Z0 * M (Hermitian). One tiny block.
// ---------------------------------------------------------------------------
__global__ __launch_bounds__(256) void vqc_build_H(const float* __restrict__ w,
                                                   float* __restrict__ Hr,
                                                   float* __restrict__ Hi) {
  __shared__ float Mr[DIM * DIM]; // Mr[col*16 + row] = Re M[row][col]
  __shared__ float Mi[DIM * DIM];
  const int tid = threadIdx.x;
  if (tid < DIM) {
    float* cr = &Mr[tid * DIM];
    float* ci = &Mi[tid * DIM];
    for (int p = 0; p < DIM; ++p) { cr[p] = (p == tid) ? 1.f : 0.f; ci[p] = 0.f; }
    for (int l = 0; l < NL; ++l) {
      // Rot(phi,theta,omega) on each wire
      for (int q = 0; q < NQ; ++q) {
        const float ph = w[(l * NQ + q) * 3 + 0];
        const float th = w[(l * NQ + q) * 3 + 1];
        const float om = w[(l * NQ + q) * 3 + 2];
        const float ct = cosf(0.5f * th), st = sinf(0.5f * th);
        const float ca = cosf(0.5f * (ph + om)), sa = sinf(0.5f * (ph + om));
        const float cb = cosf(0.5f * (ph - om)), sb = sinf(0.5f * (ph - om));
        const float u00r =  ct * ca, u00i = -ct * sa;   // epo*ct
        const float u01r = -st * cb, u01i = -st * sb;   // -conj(emo)*st
        const float u10r =  st * cb, u10i = -st * sb;   // emo*st
        const float u11r =  ct * ca, u11i =  ct * sa;   // conj(epo)*ct
        const int bp = 3 - q;          // wire q lives at bit (3-q)
        const int mask = 1 << bp;
        for (int i = 0; i < 8; ++i) {
          const int j0 = ((i >> bp) << (bp + 1)) | (i & (mask - 1));
          const int j1 = j0 | mask;
          const float a0r = cr[j0], a0i = ci[j0], a1r = cr[j1], a1i = ci[j1];
          cr[j0] = u00r * a0r - u00i * a0i + u01r * a1r - u01i * a1i;
          ci[j0] = u00r * a0i + u00i * a0r + u01r * a1i + u01i * a1r;
          cr[j1] = u10r * a0r - u10i * a0i + u11r * a1r - u11i * a1i;
          ci[j1] = u10r * a0i + u10i * a0r + u11r * a1i + u11i * a1r;
        }
      }
      // CNOT ring (c=q, t=(q+1)%4)
      for (int q = 0; q < NQ; ++q) {
        const int cbp = 3 - q, tbp = 3 - ((q + 1) % NQ);
        for (int j = 0; j < DIM; ++j) {
          if (((j >> cbp) & 1) && !((j >> tbp) & 1)) {
            const int j1 = j | (1 << tbp);
            float t0 = cr[j]; cr[j] = cr[j1]; cr[j1] = t0;
            float t1 = ci[j]; ci[j] = ci[j1]; ci[j1] = t1;
          }
        }
      }
    }
  }
  __syncthreads();
  // H[j][k] = sum_p z_p * conj(M[p][j]) * M[p][k],  z_p = +1/-1 by qubit-0 bit
  const int j = tid >> 4, k = tid & 15;
  float hr = 0.f, hi = 0.f;
  for (int p = 0; p < DIM; ++p) {
    const float z = ((p >> 3) & 1) ? -1.f : 1.f;
    const float ajr = Mr[j * DIM + p], aji = Mi[j * DIM + p];
    const float bkr = Mr[k * DIM + p], bki = Mi[k * DIM + p];
    hr += z * (ajr * bkr + aji * bki);
    hi += z * (ajr * bki - aji * bkr);
  }
  Hr[j * DIM + k] = hr;
  Hi[j * DIM + k] = hi;
}

// psi_k = (-i)^popc(k) * prod_w (bit_w(k) ? sin : cos), wire w <-> bit (3-w)
__device__ __forceinline__ void psi_amp(int k, const float* c, const float* s,
                                        float& pr, float& pi) {
  const float m = ((k & 8) ? s[0] : c[0]) * ((k & 4) ? s[1] : c[1]) *
                  ((k & 2) ? s[2] : c[2]) * ((k & 1) ? s[3] : c[3]);
  const int p = __popc(k) & 3;
  pr = (p == 0) ? m : ((p == 2) ? -m : 0.f);
  pi = (p == 1) ? -m : ((p == 3) ? m : 0.f);
}

// ---------------------------------------------------------------------------
// Main: each wave handles TPW tiles of 16 samples via 16x16 f32 WMMA.
// Y = H * P (complex, K=16 chained as 4x 16x16x4), expval = sum_j Re(P* . Y).
// ---------------------------------------------------------------------------
__global__ __launch_bounds__(256) void vqc_expval(const float* __restrict__ x,
                                                  const float* __restrict__ Hr,
                                                  const float* __restrict__ Hi,
                                                  float* __restrict__ out,
                                                  int Btot) {
  const int lane = threadIdx.x & 31;
  const int wave = threadIdx.x >> 5;
  const int half = lane >> 4;   // 0: lanes 0-15, 1: lanes 16-31
  const int l16  = lane & 15;

  // A-matrix fragments (ISA 32-bit A 16x4 layout): reg.x=H[M][k0], reg.y=H[M][k0+1]
  // with M = lane%16, k0 = chunk*4 + (lane<16 ? 0 : 2).
  v2f ar[4], ai[4];
#pragma unroll
  for (int q = 0; q < 4; ++q) {
    const int off = l16 * DIM + q * 4 + half * 2;
    ar[q] = *(const v2f*)(Hr + off);
    ai[q] = *(const v2f*)(Hi + off);
  }

  const long gwave = (long)blockIdx.x * WPB + wave;
#pragma unroll 1
  for (int t = 0; t < TPW; ++t) {
    const int s0 = (int)((gwave * TPW + t) * 16);
    int s = s0 + l16;
    if (s >= Btot) s = Btot - 1;              // clamp loads; stores guarded below
    const float4 xv = *(const float4*)(x + 4l * s);
    float c[4], sn[4];
    __sincosf(0.5f * xv.x, &sn[0], &c[0]);
    __sincosf(0.5f * xv.y, &sn[1], &c[1]);
    __sincosf(0.5f * xv.z, &sn[2], &c[2]);
    __sincosf(0.5f * xv.w, &sn[3], &c[3]);

    // B-matrix fragments of P (16 basis x 16 samples): lane holds column n=l16,
    // rows k0 / k0+1 with k0 = chunk*4 + half*2 (mirrors the A layout).
    v2f pr[4], pi[4], pin[4];
#pragma unroll
    for (int q = 0; q < 4; ++q) {
      const int k0 = q * 4 + half * 2;
      float r0, i0, r1, i1;
      psi_amp(k0,     c, sn, r0, i0);
      psi_amp(k0 + 1, c, sn, r1, i1);
      pr[q].x  = r0;  pr[q].y  = r1;
      pi[q].x  = i0;  pi[q].y  = i1;
      pin[q].x = -i0; pin[q].y = -i1;
    }

    // Yr = Hr*Pr - Hi*Pi ; Yi = Hi*Pr + Hr*Pi  (16 chained WMMAs)
    v8f yr = {}, yi = {};
#pragma unroll
    for (int q = 0; q < 4; ++q) {
      yr = __builtin_amdgcn_wmma_f32_16x16x4_f32(false, ar[q], false, pr[q],
                                                 (short)0, yr, false, false);
      yr = __builtin_amdgcn_wmma_f32_16x16x4_f32(false, ai[q], false, pin[q],
                                                 (short)0, yr, false, false);
      yi = __builtin_amdgcn_wmma_f32_16x16x4_f32(false, ai[q], false, pr[q],
                                                 (short)0, yi, false, false);
      yi = __builtin_amdgcn_wmma_f32_16x16x4_f32(false, ar[q], false, pi[q],
                                                 (short)0, yi, false, false);
    }

    // expval = sum_j (Pr*Yr + Pi*Yi); C/D layout: reg r holds row M=r+half*8,
    // column N=l16. Recompute the needed psi rows (cheap) and reduce.
    float part = 0.f;
#pragma unroll
    for (int r = 0; r < 8; ++r) {
      const int j = half * 8 + r;
      float qr, qi;
      psi_amp(j, c, sn, qr, qi);
      part += qr * yr[r] + qi * yi[r];
    }
    const float tot = part + __shfl_xor(part, 16, 32);
    if (half == 0 && (s0 + l16) < Btot) out[s0 + l16] = tot;
  }
}

extern "C" void kernel_launch(void* const* d_in, const int* in_sizes, int n_in,
                              void* d_out, int out_size, void* d_ws, size_t ws_size,
                              hipStream_t stream) {
  const float* x = (const float*)d_in[0];      // (B, 4) float32
  const float* w = (const float*)d_in[1];      // (8, 4, 3) float32
  float* out = (float*)d_out;                  // (B,) float32
  float* Hr = (float*)d_ws;                    // 256 floats
  float* Hi = Hr + DIM * DIM;                  // 256 floats
  const int B = in_sizes[0] / NQ;

  vqc_build_H<<<1, 256, 0, stream>>>(w, Hr, Hi);

  const int samples_per_block = WPB * TPW * 16;          // 1024
  const int grid = (B + samples_per_block - 1) / samples_per_block;
  vqc_expval<<<grid, 256, 0, stream>>>(x, Hr, Hi, out, B);
}